// LGCNEncoder_16922171146996
// MI455X (gfx1250) — compile-verified
//
#include <hip/hip_runtime.h>
#include <cstdint>
#include <cstddef>

#define DD 64
#define TILE 256          // edges staged per buffer (== blockDim.x)
#define NTILE 4           // tiles per block (double-buffered)
#define THREADS 256
#define NUM_LAYERS 3

// ---------------------------------------------------------------------------
// CDNA5 async global->LDS copy (ASYNCcnt-tracked). vdst = LDS byte offset,
// vaddr = 64-bit global address, "off" = no SADDR.
// ---------------------------------------------------------------------------
__device__ __forceinline__ void async_load_b32(uint32_t lds_off, const void* gaddr) {
    asm volatile("global_load_async_to_lds_b32 %0, %1, off"
                 :: "v"(lds_off), "v"(gaddr)
                 : "memory");
}

template <int N>
__device__ __forceinline__ void wait_asynccnt() {
    asm volatile("s_wait_asynccnt %0" :: "i"(N) : "memory");
}

// Device-scope hardware fp32 atomic add (no return -> STOREcnt, RMW in L2).
__device__ __forceinline__ void atomic_fadd(float* addr, float v) {
    asm volatile("global_atomic_add_f32 %0, %1, off scope:SCOPE_DEV"
                 :: "v"(addr), "v"(v)
                 : "memory");
}

__device__ __forceinline__ uint32_t lds_off_of(const void* p) {
    // generic pointer for __shared__: low 32 bits are the LDS byte offset
    return (uint32_t)(uintptr_t)p;
}

// ---------------------------------------------------------------------------
// h0[0 : n_user_f) = user_embedding ; h0[n_user_f : n_user_f+n_item_f) = item
// ---------------------------------------------------------------------------
__global__ __launch_bounds__(THREADS) void concat_kernel(
    const float* __restrict__ ue, const float* __restrict__ ie,
    float* __restrict__ h, int n_user_f, int n_total_f) {
    int i = (blockIdx.x * blockDim.x + threadIdx.x) * 4;
    if (i < n_user_f) {
        *(float4*)(h + i) = *(const float4*)(ue + i);
    } else if (i < n_total_f) {
        *(float4*)(h + i) = *(const float4*)(ie + (i - n_user_f));
    }
}

__global__ __launch_bounds__(THREADS) void zero_kernel(float* __restrict__ p, int n) {
    int i = (blockIdx.x * blockDim.x + threadIdx.x) * 4;
    if (i < n) *(float4*)(p + i) = make_float4(0.f, 0.f, 0.f, 0.f);
}

// ---------------------------------------------------------------------------
// NTILE tiles of TILE edges per block, double-buffered through LDS with the
// async copy unit. 16 lanes per edge: float4 gather of h[src] row (256B row
// per 16 lanes, L2-resident), 4x global_atomic_add_f32 into hn[dst] row.
// ---------------------------------------------------------------------------
__global__ __launch_bounds__(THREADS) void scatter_kernel(
    const float* __restrict__ h, float* __restrict__ hn,
    const int* __restrict__ src, const int* __restrict__ dst,
    const float* __restrict__ w, int E) {
    __shared__ int   s_src[2][TILE];
    __shared__ int   s_dst[2][TILE];
    __shared__ float s_w[2][TILE];

    const int tid      = threadIdx.x;
    const int blk_base = blockIdx.x * (TILE * NTILE);

    auto stage = [&](int t, int b) {
        const int i = blk_base + t * TILE + tid;
        if (i < E) {
            async_load_b32(lds_off_of(&s_src[b][tid]), src + i);
            async_load_b32(lds_off_of(&s_dst[b][tid]), dst + i);
            async_load_b32(lds_off_of(&s_w[b][tid]),   w   + i);
        }
    };

    stage(0, 0);

    const int e0   = tid >> 4;      // 0..15: edge slot
    const int lane = tid & 15;      // 0..15: quad of columns
    const int dcol = lane * 4;

    for (int t = 0; t < NTILE; ++t) {
        const int tbase = blk_base + t * TILE;
        if (tbase >= E) break;      // uniform across the block
        const int b = t & 1;

        if (t + 1 < NTILE) {
            stage(t + 1, b ^ 1);    // overlap next tile's copy with this tile
            wait_asynccnt<3>();     // async loads complete in order: tile t done
        } else {
            wait_asynccnt<0>();
        }
        __syncthreads();            // all waves' staged data visible

        const int tile_n = (E - tbase < TILE) ? (E - tbase) : TILE;
        for (int e = e0; e < tile_n; e += 16) {
            const int en = e + 16;  // prefetch next source row
            if (en < tile_n && lane == 0) {
                __builtin_prefetch(h + (size_t)s_src[b][en] * DD, 0, 3);
            }
            const int   s   = s_src[b][e];
            const int   d   = s_dst[b][e];
            const float wgt = s_w[b][e];

            const float4 v = *(const float4*)(h + (size_t)s * DD + dcol);
            float* o = hn + (size_t)d * DD + dcol;
            atomic_fadd(o + 0, v.x * wgt);
            atomic_fadd(o + 1, v.y * wgt);
            atomic_fadd(o + 2, v.z * wgt);
            atomic_fadd(o + 3, v.w * wgt);
        }
        __syncthreads();            // buffer b is free to be restaged next iter
    }
}

// ---------------------------------------------------------------------------
// out[row] (ov? = : +=) scale * h[node(row)], rows = [uid rows | iid rows]
// ---------------------------------------------------------------------------
__global__ __launch_bounds__(THREADS) void gather_kernel(
    const float* __restrict__ h,
    const int* __restrict__ uid, const int* __restrict__ iid,
    float* __restrict__ out, int batch, int n_users, float scale, int overwrite) {
    const int g    = blockIdx.x * blockDim.x + threadIdx.x;
    const int row  = g >> 4;
    const int dcol = (g & 15) * 4;
    if (row >= 2 * batch) return;
    const int node = (row < batch) ? uid[row] : (n_users + iid[row - batch]);
    const float4 v = *(const float4*)(h + (size_t)node * DD + dcol);
    float* o = out + (size_t)row * DD + dcol;
    if (overwrite) {
        o[0] = v.x * scale; o[1] = v.y * scale;
        o[2] = v.z * scale; o[3] = v.w * scale;
    } else {
        o[0] += v.x * scale; o[1] += v.y * scale;
        o[2] += v.z * scale; o[3] += v.w * scale;
    }
}

extern "C" void kernel_launch(void* const* d_in, const int* in_sizes, int n_in,
                              void* d_out, int out_size, void* d_ws, size_t ws_size,
                              hipStream_t stream) {
    const float* ue   = (const float*)d_in[0];   // [N_USERS, 64]
    const float* ie   = (const float*)d_in[1];   // [N_ITEMS, 64]
    const float* ew   = (const float*)d_in[2];   // [E]
    const int*   eidx = (const int*)d_in[3];     // [2, E]
    const int*   uid  = (const int*)d_in[4];     // [BATCH]
    const int*   iid  = (const int*)d_in[5];     // [BATCH]

    const int n_user_f = in_sizes[0];
    const int n_item_f = in_sizes[1];
    const int E        = in_sizes[2];
    const int batch    = in_sizes[4];
    const int n_users  = n_user_f / DD;
    const int n_total  = n_user_f + n_item_f;    // N_NODES * 64 floats

    const int* src = eidx;
    const int* dst = eidx + E;

    float* h0  = (float*)d_ws;                   // ping (38.4 MB)
    float* h1  = h0 + n_total;                   // pong (38.4 MB)
    float* out = (float*)d_out;

    const float scale = 1.0f / (float)(NUM_LAYERS + 1);

    const int copy_blocks   = (n_total / 4 + THREADS - 1) / THREADS;
    const int gather_blocks = (2 * batch * 16 + THREADS - 1) / THREADS;
    const int epb           = TILE * NTILE;      // edges per block
    const int edge_blocks   = (E + epb - 1) / epb;

    // layer 0: x itself (overwrite the poisoned d_out)
    concat_kernel<<<copy_blocks, THREADS, 0, stream>>>(ue, ie, h0, n_user_f, n_total);
    gather_kernel<<<gather_blocks, THREADS, 0, stream>>>(h0, uid, iid, out, batch,
                                                         n_users, scale, 1);

    float* hc = h0;
    float* hn = h1;
    for (int l = 0; l < NUM_LAYERS; ++l) {
        zero_kernel<<<copy_blocks, THREADS, 0, stream>>>(hn, n_total);
        scatter_kernel<<<edge_blocks, THREADS, 0, stream>>>(hc, hn, src, dst, ew, E);
        gather_kernel<<<gather_blocks, THREADS, 0, stream>>>(hn, uid, iid, out, batch,
                                                             n_users, scale, 0);
        float* t = hc; hc = hn; hn = t;
    }
}